// ClimateGNN_12687333392439
// MI455X (gfx1250) — compile-verified
//
#include <hip/hip_runtime.h>
#include <hip/hip_bf16.h>

typedef float v2f __attribute__((ext_vector_type(2)));
typedef float v8f __attribute__((ext_vector_type(8)));

#define FH 64  // feature width for layers 1/2 (F_IN == H == 64)

// ---------------------------------------------------------------- utilities
__global__ void gnn_zero_f32(float* __restrict__ p, int n) {
    int i = blockIdx.x * blockDim.x + threadIdx.x;
    if (i < n) p[i] = 0.0f;
}

// ------------------------------------------------------------ edge pipeline
// ew = exp(-(d/200)^2); int64 -> int32 edge indices; deg[dst] += ew
__global__ void gnn_edge_prep(const long long* __restrict__ ei,
                              const float* __restrict__ attr,
                              int* __restrict__ s32, int* __restrict__ d32,
                              float* __restrict__ ew, float* __restrict__ deg, int E) {
    int e = blockIdx.x * blockDim.x + threadIdx.x;
    if (e >= E) return;
    int s = (int)ei[e];
    int d = (int)ei[(size_t)E + e];
    float a = attr[e] * 0.005f;            // 1/200
    float w = __expf(-a * a);
    s32[e] = s;
    d32[e] = d;
    ew[e]  = w;
    atomicAdd(&deg[d], w);
}

// dinv = rsqrt(deg + 1)  (in place)
__global__ void gnn_dinv(float* __restrict__ deg, int n) {
    int i = blockIdx.x * blockDim.x + threadIdx.x;
    if (i < n) deg[i] = rsqrtf(deg[i] + 1.0f);
}

// norm[e] = dinv[src] * ew[e] * dinv[dst]  (in place on ew buffer)
__global__ void gnn_norm(float* __restrict__ nrm, const int* __restrict__ s32,
                         const int* __restrict__ d32, const float* __restrict__ dinv, int E) {
    int e = blockIdx.x * blockDim.x + threadIdx.x;
    if (e >= E) return;
    nrm[e] = dinv[s32[e]] * nrm[e] * dinv[d32[e]];
}

// ---------------------------------------------------------------- f32 WMMA GEMM
// O[nrows x 64] = X[nrows x 64] @ W[64 x 64]; one wave per 16x16 tile.
// nrows must be a multiple of 16 (100000 = 6250*16).
__global__ void gnn_gemm64_wmma(const float* __restrict__ X, const float* __restrict__ W,
                                float* __restrict__ O, int nrows) {
    int wave = (blockIdx.x * blockDim.x + threadIdx.x) >> 5;
    int lane = threadIdx.x & 31;
    int tm = wave >> 2;          // row tile (16 nodes)
    int tc = wave & 3;           // col tile (16 of 64 outputs)
    if (tm * 16 >= nrows) return;

    int half = lane >> 4;        // 0: K even pair / M 0..7 ; 1: K odd pair / M 8..15
    int l16  = lane & 15;

    const float* xrow = X + (size_t)(tm * 16 + l16) * FH;   // A: M = l16
    const float* wcol = W + tc * 16 + l16;                  // B: N = l16

    v8f acc = {};
#pragma unroll
    for (int kk = 0; kk < FH; kk += 4) {
        v2f a, b;
        // A 16x4 layout: lanes 0-15 hold K=kk,kk+1; lanes 16-31 hold K=kk+2,kk+3
        a.x = xrow[kk + 2 * half];
        a.y = xrow[kk + 2 * half + 1];
        // B 4x16 layout mirrors A with N striped across lanes
        b.x = wcol[(size_t)(kk + 2 * half) * FH];
        b.y = wcol[(size_t)(kk + 2 * half + 1) * FH];
        acc = __builtin_amdgcn_wmma_f32_16x16x4_f32(
            /*neg_a=*/false, a, /*neg_b=*/false, b,
            /*c_mod=*/(short)0, acc, /*reuse_a=*/false, /*reuse_b=*/false);
    }
    // D layout: VGPR r -> M = r (lanes 0-15), M = 8+r (lanes 16-31)
    float* orow = O + (size_t)(tm * 16 + 8 * half) * FH + tc * 16 + l16;
#pragma unroll
    for (int r = 0; r < 8; ++r)
        orow[(size_t)r * FH] = acc[r];
}

// ------------------------------------------------------- edge gather-scatter
// One wave per edge: lane handles features lane and lane+32. L2-resident atomics.
__global__ void gnn_edge_agg(float* __restrict__ agg, const float* __restrict__ xw,
                             const int* __restrict__ s32, const int* __restrict__ d32,
                             const float* __restrict__ nrm, int E) {
    int wave = (blockIdx.x * blockDim.x + threadIdx.x) >> 5;
    int lane = threadIdx.x & 31;
    if (wave >= E) return;
    int s = s32[wave];
    int d = d32[wave];
    float nv = nrm[wave];
    const float* xs = xw + (size_t)s * FH;
    float* ad = agg + (size_t)d * FH;
    atomicAdd(&ad[lane],      nv * xs[lane]);
    atomicAdd(&ad[lane + 32], nv * xs[lane + 32]);
}

// agg = [relu](agg + dinv^2 * xw + bias)   (fused, coalesced)
__global__ void gnn_selfloop_bias_act(float* __restrict__ agg, const float* __restrict__ xw,
                                      const float* __restrict__ dinv,
                                      const float* __restrict__ bias, int relu, int total) {
    int idx = blockIdx.x * blockDim.x + threadIdx.x;
    if (idx >= total) return;
    int n = idx >> 6;
    int f = idx & 63;
    float di = dinv[n];
    float v = agg[idx] + di * di * xw[idx] + bias[f];
    if (relu) v = fmaxf(v, 0.0f);
    agg[idx] = v;
}

// ------------------------------------------------------------- layer 3 (H->1)
// One wave per node: 64-wide dot with W3, shuffle reduction.
__global__ void gnn_dot64(const float* __restrict__ h, const float* __restrict__ w3,
                          float* __restrict__ o, int n) {
    int wave = (blockIdx.x * blockDim.x + threadIdx.x) >> 5;
    int lane = threadIdx.x & 31;
    if (wave >= n) return;
    const float* hr = h + (size_t)wave * FH;
    float v = hr[lane] * w3[lane] + hr[lane + 32] * w3[lane + 32];
    for (int off = 16; off > 0; off >>= 1) v += __shfl_down(v, off);
    if (lane == 0) o[wave] = v;
}

__global__ void gnn_edge_agg_scalar(float* __restrict__ o, const float* __restrict__ xw3,
                                    const int* __restrict__ s32, const int* __restrict__ d32,
                                    const float* __restrict__ nrm, int E) {
    int e = blockIdx.x * blockDim.x + threadIdx.x;
    if (e >= E) return;
    atomicAdd(&o[d32[e]], nrm[e] * xw3[s32[e]]);
}

__global__ void gnn_final(float* __restrict__ o, const float* __restrict__ xw3,
                          const float* __restrict__ dinv, const float* __restrict__ b3, int n) {
    int i = blockIdx.x * blockDim.x + threadIdx.x;
    if (i >= n) return;
    float di = dinv[i];
    o[i] += di * di * xw3[i] + b3[0];
}

// ---------------------------------------------------------------- launcher
extern "C" void kernel_launch(void* const* d_in, const int* in_sizes, int n_in,
                              void* d_out, int out_size, void* d_ws, size_t ws_size,
                              hipStream_t stream) {
    const float*     x    = (const float*)d_in[0];
    const long long* ei   = (const long long*)d_in[1];
    const float*     attr = (const float*)d_in[2];
    const float*     W1   = (const float*)d_in[3];
    const float*     b1   = (const float*)d_in[4];
    const float*     W2   = (const float*)d_in[5];
    const float*     b2   = (const float*)d_in[6];
    const float*     W3   = (const float*)d_in[7];
    const float*     b3   = (const float*)d_in[8];
    float* out = (float*)d_out;

    const int N = in_sizes[0] / FH;   // 100000
    const int E = in_sizes[2];        // 1600000
    const int NH = N * FH;

    // Workspace layout (all 4-byte typed, offsets are multiples of 64B)
    char* ws = (char*)d_ws;
    int*   s32  = (int*)  ws;                                     // E ints
    int*   d32  = (int*)  (ws + (size_t)E * 4);                   // E ints
    float* nrm  = (float*)(ws + (size_t)E * 8);                   // E floats (ew -> norm)
    float* dinv = (float*)(ws + (size_t)E * 12);                  // N floats (deg -> dinv)
    float* xw3  = (float*)(ws + (size_t)E * 12 + (size_t)N * 4);  // N floats
    float* bufA = (float*)(ws + (size_t)E * 12 + (size_t)N * 8);  // N*64 floats
    float* bufB = bufA + (size_t)NH;                              // N*64 floats

    const int TPB = 256;
    dim3 blk(TPB);
    int gN   = (N  + TPB - 1) / TPB;
    int gE   = (E  + TPB - 1) / TPB;
    int gNH  = (NH + TPB - 1) / TPB;
    int gEw  = (E + 7) / 8;                       // 8 waves/block, 1 edge/wave
    int gNw  = (N + 7) / 8;                       // 1 node/wave
    int tiles = (N / 16) * 4;                     // 16x16 tiles, 4 col tiles
    int gGemm = (tiles + 7) / 8;                  // 8 waves/block

    // ---- edge preprocessing (once, reused by all 3 layers)
    gnn_zero_f32<<<gN, blk, 0, stream>>>(dinv, N);                       // deg = 0
    gnn_edge_prep<<<gE, blk, 0, stream>>>(ei, attr, s32, d32, nrm, dinv, E);
    gnn_dinv<<<gN, blk, 0, stream>>>(dinv, N);
    gnn_norm<<<gE, blk, 0, stream>>>(nrm, s32, d32, dinv, E);

    // ---- layer 1: h1 = relu(gcn(x, W1, b1))  -> bufB
    gnn_gemm64_wmma<<<gGemm, blk, 0, stream>>>(x, W1, bufA, N);
    gnn_zero_f32<<<gNH, blk, 0, stream>>>(bufB, NH);
    gnn_edge_agg<<<gEw, blk, 0, stream>>>(bufB, bufA, s32, d32, nrm, E);
    gnn_selfloop_bias_act<<<gNH, blk, 0, stream>>>(bufB, bufA, dinv, b1, 1, NH);

    // ---- layer 2: h2 = relu(gcn(h1, W2, b2)) -> bufB (ping-pong via bufA)
    gnn_gemm64_wmma<<<gGemm, blk, 0, stream>>>(bufB, W2, bufA, N);
    gnn_zero_f32<<<gNH, blk, 0, stream>>>(bufB, NH);
    gnn_edge_agg<<<gEw, blk, 0, stream>>>(bufB, bufA, s32, d32, nrm, E);
    gnn_selfloop_bias_act<<<gNH, blk, 0, stream>>>(bufB, bufA, dinv, b2, 1, NH);

    // ---- layer 3: out = gcn(h2, W3, b3)[:, 0]
    gnn_dot64<<<gNw, blk, 0, stream>>>(bufB, W3, xw3, N);
    gnn_zero_f32<<<gN, blk, 0, stream>>>(out, N);
    gnn_edge_agg_scalar<<<gE, blk, 0, stream>>>(out, xw3, s32, d32, nrm, E);
    gnn_final<<<gN, blk, 0, stream>>>(out, xw3, dinv, b3, N);
}